// MLPRegressor_51221779972563
// MI455X (gfx1250) — compile-verified
//
#include <hip/hip_runtime.h>

typedef __attribute__((ext_vector_type(2))) float v2f;
typedef __attribute__((ext_vector_type(8))) float v8f;

#define SEQ  256
#define DM   32
#define XDIM 128
#define HIDN 64

static __device__ inline v8f wmma4(v2f a, v2f b, v8f c) {
  // D = A(16x4 f32) * B(4x16 f32) + C(16x16 f32), wave32
  return __builtin_amdgcn_wmma_f32_16x16x4_f32(
      /*neg_a=*/false, a, /*neg_b=*/false, b,
      /*c_mod=*/(short)0, c, /*reuse_a=*/false, /*reuse_b=*/false);
}

// ---------------------------------------------------------------------------
// Kernel 1: per-batch-row masked pooling of {emb_p/5, emb_c/2, relu_p, relu_c}
// Pooling is linear, so the 32x32 matmuls (Wp2/Wc2) are deferred to kernel 2.
// Block = 256 threads (8 waves) per b. lane = feature dim (0..31), so each
// embedding-row gather is one coalesced 128B load across the wave.
// ---------------------------------------------------------------------------
__global__ __launch_bounds__(256) void pool_kernel(
    const float* __restrict__ cont_p, const float* __restrict__ cont_c,
    const int*   __restrict__ cat_p,  const int*   __restrict__ cat_c,
    const int*   __restrict__ lens,
    const float* __restrict__ Wp1, const float* __restrict__ bp1,
    const float* __restrict__ Wc1, const float* __restrict__ bc1,
    const float* __restrict__ E_gender, const float* __restrict__ E_korean,
    const float* __restrict__ E_primary, const float* __restrict__ E_job,
    const float* __restrict__ E_rep, const float* __restrict__ E_place,
    const float* __restrict__ E_add,
    float* __restrict__ ws)
{
  __shared__ float red[8][4][32];
  const int b    = blockIdx.x;
  const int lane = threadIdx.x & 31;
  const int wave = threadIdx.x >> 5;
  const int len  = lens[b];                  // 1..S

  // per-lane (per-dim) weights, stay in VGPRs
  const float w0 = Wp1[0*DM + lane], w1 = Wp1[1*DM + lane], w2 = Wp1[2*DM + lane];
  const float u0 = Wc1[0*DM + lane], u1 = Wc1[1*DM + lane];
  const float bP = bp1[lane],        bC = bc1[lane];

  float s_rp = 0.f, s_rc = 0.f, s_ep = 0.f, s_ec = 0.f;

  for (int s = wave; s < len; s += 8) {
    const long t = (long)b * SEQ + s;
    const float* cp = cont_p + t * 3;        // wave-uniform scalars -> SMEM
    const float* cc = cont_c + t * 2;
    const int*   ip = cat_p  + t * 5;
    const int*   ic = cat_c  + t * 2;

    float rp = fmaxf(fmaf(cp[2], w2, fmaf(cp[1], w1, fmaf(cp[0], w0, bP))), 0.f);
    float rc = fmaxf(fmaf(cc[1], u1, fmaf(cc[0], u0, bC)), 0.f);
    s_rp += rp;
    s_rc += rc;

    s_ep += E_gender [ip[0]*DM + lane] + E_korean[ip[1]*DM + lane]
          + E_primary[ip[2]*DM + lane] + E_job   [ip[3]*DM + lane]
          + E_place  [ic[0]*DM + lane];
    s_ec += E_add    [ic[1]*DM + lane] + E_rep   [ip[4]*DM + lane];
  }

  red[wave][0][lane] = s_ep;
  red[wave][1][lane] = s_ec;
  red[wave][2][lane] = s_rp;
  red[wave][3][lane] = s_rc;
  __syncthreads();

  if (threadIdx.x < 128) {
    const int g = threadIdx.x >> 5, d = threadIdx.x & 31;
    float acc = 0.f;
#pragma unroll
    for (int w = 0; w < 8; ++w) acc += red[w][g][d];
    const float inv = 1.0f / (float)len;
    if (g == 0)      acc *= 0.2f * inv;      // catp = (a1..a5)/5, pooled
    else if (g == 1) acc *= 0.5f * inv;      // catc = (a6+a7)/2, pooled
    else             acc *= inv;             // pooled relu activations
    ws[(long)b * XDIM + g * 32 + d] = acc;
  }
}

// ---------------------------------------------------------------------------
// Kernel 2: one wave per 16 batch rows. All matmuls via V_WMMA_F32_16X16X4_F32.
//  cp = (pooled_relu_p @ Wp2) + bp2 ; cc likewise          (K=32, 2 n-tiles)
//  x  = [catp | catc | cp | cc]  (16x128, cols 0..63 straight from ws)
//  h  = relu(x @ W1 + b1)                                   (K=128, 4 n-tiles)
//  out= relu(h @ W2 + b2)                                   (64->2, VALU)
// Operand layouts per CDNA5 ISA 7.12.2:
//  A: lane m=lane&15, a.x=A[m][4i+2*(lane>>4)], a.y=+1
//  B: lane n=lane&15, b.x=B[4i+2*(lane>>4)][n], b.y=+1
//  C/D: vreg r -> row r+8*(lane>>4), col lane&15
// ---------------------------------------------------------------------------
__global__ __launch_bounds__(32) void mlp_wmma_kernel(
    const float* __restrict__ ws,
    const float* __restrict__ Wp2, const float* __restrict__ bp2,
    const float* __restrict__ Wc2, const float* __restrict__ bc2,
    const float* __restrict__ W1,  const float* __restrict__ b1,
    const float* __restrict__ W2,  const float* __restrict__ b2,
    float* __restrict__ out)
{
  __shared__ float xt2[16][64];   // cols 64..127 of x tile (cp | cc)
  __shared__ float ht[16][64];    // hidden tile

  const int lane  = threadIdx.x;
  const int mrow  = lane & 15;
  const int khalf = lane >> 4;               // 0: K+0/1, 1: K+2/3
  const int rb    = blockIdx.x * 16;

  // ---- stage A: cp = Sp @ Wp2, cc = Sc @ Wc2 (16x32 each) ----
  v8f ap0 = {}, ap1 = {}, ac0 = {}, ac1 = {};
  const float* rowp = ws + (long)(rb + mrow) * XDIM + 64;   // pooled relu_p
  const float* rowc = rowp + 32;                            // pooled relu_c
  for (int i = 0; i < 8; ++i) {
    const int k = 4 * i + 2 * khalf;
    v2f a_p; a_p.x = rowp[k]; a_p.y = rowp[k + 1];
    v2f a_c; a_c.x = rowc[k]; a_c.y = rowc[k + 1];
    v2f bp0v; bp0v.x = Wp2[k * DM + mrow];       bp0v.y = Wp2[(k + 1) * DM + mrow];
    v2f bp1v; bp1v.x = Wp2[k * DM + 16 + mrow];  bp1v.y = Wp2[(k + 1) * DM + 16 + mrow];
    v2f bc0v; bc0v.x = Wc2[k * DM + mrow];       bc0v.y = Wc2[(k + 1) * DM + mrow];
    v2f bc1v; bc1v.x = Wc2[k * DM + 16 + mrow];  bc1v.y = Wc2[(k + 1) * DM + 16 + mrow];
    ap0 = wmma4(a_p, bp0v, ap0);
    ap1 = wmma4(a_p, bp1v, ap1);
    ac0 = wmma4(a_c, bc0v, ac0);
    ac1 = wmma4(a_c, bc1v, ac1);
  }
#pragma unroll
  for (int r = 0; r < 8; ++r) {
    const int row = r + 8 * khalf;
    xt2[row][     mrow] = ap0[r] + bp2[mrow];
    xt2[row][16 + mrow] = ap1[r] + bp2[16 + mrow];
    xt2[row][32 + mrow] = ac0[r] + bc2[mrow];
    xt2[row][48 + mrow] = ac1[r] + bc2[16 + mrow];
  }
  __syncthreads();

  // ---- stage B: h = x(16x128) @ W1(128x64) ----
  v8f hacc[4] = {{}, {}, {}, {}};
  const float* rowx = ws + (long)(rb + mrow) * XDIM;
  for (int i = 0; i < 16; ++i) {             // K = 0..63 from ws (catp|catc)
    const int k = 4 * i + 2 * khalf;
    v2f a; a.x = rowx[k]; a.y = rowx[k + 1];
#pragma unroll
    for (int n = 0; n < 4; ++n) {
      v2f bv;
      bv.x = W1[k * HIDN + n * 16 + mrow];
      bv.y = W1[(k + 1) * HIDN + n * 16 + mrow];
      hacc[n] = wmma4(a, bv, hacc[n]);
    }
  }
  for (int i = 16; i < 32; ++i) {            // K = 64..127 from LDS (cp|cc)
    const int k = 4 * i + 2 * khalf;
    v2f a; a.x = xt2[mrow][k - 64]; a.y = xt2[mrow][k - 63];
#pragma unroll
    for (int n = 0; n < 4; ++n) {
      v2f bv;
      bv.x = W1[k * HIDN + n * 16 + mrow];
      bv.y = W1[(k + 1) * HIDN + n * 16 + mrow];
      hacc[n] = wmma4(a, bv, hacc[n]);
    }
  }
#pragma unroll
  for (int n = 0; n < 4; ++n)
#pragma unroll
    for (int r = 0; r < 8; ++r)
      ht[r + 8 * khalf][n * 16 + mrow] = fmaxf(hacc[n][r] + b1[n * 16 + mrow], 0.f);
  __syncthreads();

  // ---- stage C: out = relu(h @ W2 + b2), 16 rows x 2 outs = 32 lanes ----
  const int m = lane >> 1, o = lane & 1;
  float acc = b2[o];
#pragma unroll
  for (int k = 0; k < HIDN; ++k) acc = fmaf(ht[m][k], W2[k * 2 + o], acc);
  out[(long)(rb + m) * 2 + o] = fmaxf(acc, 0.f);
}

// ---------------------------------------------------------------------------
extern "C" void kernel_launch(void* const* d_in, const int* in_sizes, int n_in,
                              void* d_out, int out_size, void* d_ws, size_t ws_size,
                              hipStream_t stream) {
  const float* cont_p   = (const float*)d_in[0];
  const float* cont_c   = (const float*)d_in[1];
  const int*   cat_p    = (const int*)  d_in[2];
  const int*   cat_c    = (const int*)  d_in[3];
  const int*   lens     = (const int*)  d_in[4];
  const float* Wp1      = (const float*)d_in[5];
  const float* bp1      = (const float*)d_in[6];
  const float* Wp2      = (const float*)d_in[7];
  const float* bp2      = (const float*)d_in[8];
  const float* Wc1      = (const float*)d_in[9];
  const float* bc1      = (const float*)d_in[10];
  const float* Wc2      = (const float*)d_in[11];
  const float* bc2      = (const float*)d_in[12];
  const float* E_gender = (const float*)d_in[13];
  const float* E_korean = (const float*)d_in[14];
  const float* E_primary= (const float*)d_in[15];
  const float* E_job    = (const float*)d_in[16];
  const float* E_rep    = (const float*)d_in[17];
  const float* E_place  = (const float*)d_in[18];
  const float* E_add    = (const float*)d_in[19];
  const float* W1       = (const float*)d_in[20];
  const float* b1       = (const float*)d_in[21];
  const float* W2       = (const float*)d_in[22];
  const float* b2       = (const float*)d_in[23];

  const int B = 4096;
  float* ws = (float*)d_ws;                  // B * 128 floats = 2 MB

  pool_kernel<<<B, 256, 0, stream>>>(
      cont_p, cont_c, cat_p, cat_c, lens,
      Wp1, bp1, Wc1, bc1,
      E_gender, E_korean, E_primary, E_job, E_rep, E_place, E_add, ws);

  mlp_wmma_kernel<<<B / 16, 32, 0, stream>>>(
      ws, Wp2, bp2, Wc2, bc2, W1, b1, W2, b2, (float*)d_out);
}